// Position_Encoding_33827162423832
// MI455X (gfx1250) — compile-verified
//
#include <hip/hip_runtime.h>
#include <cstdint>
#include <cstddef>

// ---------------------------------------------------------------------------
// Positional-encoding + inverted dropout, MI455X (gfx1250, wave32).
// HBM-bound streaming kernel: 192 MiB traffic -> ~8.6 us floor @ 23.3 TB/s.
// PE (depends only on (s,d)) computed once per lane, reused over 8 batches.
// Data movement uses the CDNA5 async global->LDS DMA path (ASYNCcnt) with
// double buffering; trig uses Cody-Waite reduction + native v_sin/v_cos
// (args bounded by 4096 rad -> k <= 652, reduction error ~1e-7*k, far below
// the fp32 rounding already present in the angle itself).
// ---------------------------------------------------------------------------

typedef float f32x4 __attribute__((ext_vector_type(4)));
typedef int   v4i   __attribute__((vector_size(16)));          // matches builtin pointee
typedef __attribute__((address_space(1))) v4i gv4i;            // global (AS1)
typedef __attribute__((address_space(3))) v4i lv4i;            // LDS (AS3)

#define TPB    256
#define BATCH  8
#define SEQ    4096
#define DMODEL 512
#define PLANE  (SEQ * DMODEL)   // 2,097,152 floats per batch plane

#if defined(__AMDGCN__) &&                                              \
    __has_builtin(__builtin_amdgcn_global_load_async_to_lds_b128) &&    \
    __has_builtin(__builtin_amdgcn_s_wait_asynccnt)
#define USE_ASYNC_LDS 1
#else
#define USE_ASYNC_LDS 0
#endif

// sin/cos for |ang| < 4096 rad: Cody-Waite range reduction into [-pi, pi],
// then native hardware transcendentals (valid in that range).
__device__ __forceinline__ float fast_sin_bounded(float ang) {
    const float INV_2PI = 0.15915494309189535f;
    const float PI2_HI  = 6.2831854820251465f;   // fp32(2*pi)
    const float PI2_LO  = -1.7484555e-7f;        // 2*pi - PI2_HI
    float k = rintf(ang * INV_2PI);
    float r = __builtin_fmaf(-k, PI2_HI, ang);
    r = __builtin_fmaf(-k, PI2_LO, r);
    return __sinf(r);
}
__device__ __forceinline__ float fast_cos_bounded(float ang) {
    const float INV_2PI = 0.15915494309189535f;
    const float PI2_HI  = 6.2831854820251465f;
    const float PI2_LO  = -1.7484555e-7f;
    float k = rintf(ang * INV_2PI);
    float r = __builtin_fmaf(-k, PI2_HI, ang);
    r = __builtin_fmaf(-k, PI2_LO, r);
    return __cosf(r);
}

__global__ __launch_bounds__(TPB) void pe_dropout_kernel(
    const float* __restrict__ x,
    const float* __restrict__ mask,
    float* __restrict__ out)
{
    const int tid = threadIdx.x;
    const int e   = (blockIdx.x * TPB + tid) * 4;   // index into [S*D] plane
    if (e >= PLANE) return;

    const int srow = e >> 9;     // D = 512
    const int dcol = e & 511;    // multiple of 4 -> 4 channels share one row

    // angle = s * 10000^(-2 i / 512) = s * exp2(i * C2)
    const float C2 = -0.05190512648261504f;  // -2*log2(10000)/512
    float pe[4];
#pragma unroll
    for (int j = 0; j < 4; ++j) {
        float ang = (float)srow * exp2f((float)(dcol + j) * C2);
        pe[j] = ((j & 1) == 0) ? fast_sin_bounded(ang)     // even channel -> sin
                               : fast_cos_bounded(ang);    // odd  channel -> cos
    }

    const float P    = 0.1f;
    const float INVK = 1.0f / 0.9f;

#if USE_ASYNC_LDS
    // Lane-private double-buffered staging tiles: 2 * 2 * 256 * 16B = 16 KB LDS.
    __shared__ f32x4 sx[2][TPB];
    __shared__ f32x4 sm[2][TPB];

    auto cp16 = [&](const float* g, f32x4* l) {
        __builtin_amdgcn_global_load_async_to_lds_b128(
            (gv4i*)(uintptr_t)g,
            (lv4i*)(uintptr_t)l,
            /*offset=*/0, /*cpol=*/0);
    };

    const float* xp = x + e;
    const float* mp = mask + e;

    // Prologue: batch 0 into buffer 0.
    cp16(xp, &sx[0][tid]);
    cp16(mp, &sm[0][tid]);

#pragma unroll
    for (int b = 0; b < BATCH; ++b) {
        const int cur = b & 1;
        if (b + 1 < BATCH) {
            // Prefetch next batch plane into the other buffer, then wait
            // until only those 2 just-issued async ops remain outstanding.
            cp16(xp + (size_t)(b + 1) * PLANE, &sx[cur ^ 1][tid]);
            cp16(mp + (size_t)(b + 1) * PLANE, &sm[cur ^ 1][tid]);
            __builtin_amdgcn_s_wait_asynccnt(2);
        } else {
            __builtin_amdgcn_s_wait_asynccnt(0);
        }
        asm volatile("" ::: "memory");   // keep LDS reads below the wait

        f32x4 xv = sx[cur][tid];
        f32x4 mv = sm[cur][tid];
        f32x4 o;
        o.x = (xv.x + pe[0]) * (mv.x >= P ? INVK : 0.0f);
        o.y = (xv.y + pe[1]) * (mv.y >= P ? INVK : 0.0f);
        o.z = (xv.z + pe[2]) * (mv.z >= P ? INVK : 0.0f);
        o.w = (xv.w + pe[3]) * (mv.w >= P ? INVK : 0.0f);

        __builtin_nontemporal_store(o, (f32x4*)(out + (size_t)b * PLANE + e));
    }
#else
    // Fallback: direct nontemporal B128 streaming loads.
#pragma unroll
    for (int b = 0; b < BATCH; ++b) {
        const size_t off = (size_t)b * PLANE + (size_t)e;
        f32x4 xv = __builtin_nontemporal_load((const f32x4*)(x + off));
        f32x4 mv = __builtin_nontemporal_load((const f32x4*)(mask + off));
        f32x4 o;
        o.x = (xv.x + pe[0]) * (mv.x >= P ? INVK : 0.0f);
        o.y = (xv.y + pe[1]) * (mv.y >= P ? INVK : 0.0f);
        o.z = (xv.z + pe[2]) * (mv.z >= P ? INVK : 0.0f);
        o.w = (xv.w + pe[3]) * (mv.w >= P ? INVK : 0.0f);
        __builtin_nontemporal_store(o, (f32x4*)(out + off));
    }
#endif
}

extern "C" void kernel_launch(void* const* d_in, const int* in_sizes, int n_in,
                              void* d_out, int out_size, void* d_ws, size_t ws_size,
                              hipStream_t stream)
{
    (void)in_sizes; (void)n_in; (void)out_size; (void)d_ws; (void)ws_size;
    const float* x    = (const float*)d_in[0];
    const float* mask = (const float*)d_in[1];
    float*       out  = (float*)d_out;

    constexpr int blocks = PLANE / (TPB * 4);   // 2048 blocks, 8 wave32 each
    pe_dropout_kernel<<<blocks, TPB, 0, stream>>>(x, mask, out);
}